// QPProjectionWithSlack_20650202759696
// MI455X (gfx1250) — compile-verified
//
#include <hip/hip_runtime.h>

typedef __attribute__((ext_vector_type(2))) float v2f;
typedef __attribute__((ext_vector_type(8))) float v8f;

#define N_DIM 128       // n
#define M_DIM 64        // m
#define A_PITCH 129     // (i*129+k)%64 == (i+k)%64 -> conflict-free rows & cols
#define S_PITCH 65
#define SIGMA_C 0.1f
#define STEPF 0.99f
#define NITER 30

#if defined(__gfx1250__) && __has_builtin(__builtin_amdgcn_global_load_async_to_lds_b32)
#define HAVE_ASYNC_LDS 1
#endif

__launch_bounds__(256)
__global__ void qp_ipm_gfx1250_kernel(const float* __restrict__ xraw_g,
                                      const float* __restrict__ A_g,
                                      const float* __restrict__ b_g,
                                      float* __restrict__ out_g)
{
    __shared__ float Ash[M_DIM * A_PITCH];   // A (64 x 128), padded
    __shared__ float Ssh[M_DIM * S_PITCH];   // Schur matrix / LDL^T factor
    __shared__ float xr[N_DIM], bv[M_DIM];
    __shared__ float zx[N_DIM], zs[M_DIM];
    __shared__ float lam1[M_DIM], lam2[M_DIM], s1[M_DIM], s2[M_DIM];
    __shared__ float rp1[M_DIM], rp2[M_DIM], rds[M_DIM];
    __shared__ float rc1[M_DIM], rc2[M_DIM];
    __shared__ float w1v[M_DIM], whv[M_DIM], Dv[M_DIM];
    __shared__ float rhss[M_DIM], g1[M_DIM];
    __shared__ float rhat[N_DIM], yv[M_DIM];
    __shared__ float dzx[N_DIM], dzs[M_DIM];
    __shared__ float ds1v[M_DIM], ds2v[M_DIM], dl1[M_DIM], dl2[M_DIM];
    __shared__ float red[8];
    __shared__ float scal[2];

    const int t    = blockIdx.x;
    const int tid  = threadIdx.x;
    const int lane = tid & 31;
    const int wave = tid >> 5;

    // ---- stage A into LDS: CDNA5 async global->LDS path (ASYNCcnt) ----
    const float* Ab = A_g + (size_t)t * (M_DIM * N_DIM);
#ifdef HAVE_ASYNC_LDS
    {
        typedef __attribute__((address_space(1))) int as1_int;
        typedef __attribute__((address_space(3))) int as3_int;
        for (int idx = tid; idx < M_DIM * N_DIM; idx += 256) {
            int r = idx >> 7, c = idx & 127;
            __builtin_amdgcn_global_load_async_to_lds_b32(
                (as1_int*)(Ab + idx),
                (as3_int*)(&Ash[r * A_PITCH + c]), 0, 0);
        }
#if __has_builtin(__builtin_amdgcn_s_wait_asynccnt)
        __builtin_amdgcn_s_wait_asynccnt(0);
#else
        asm volatile("s_wait_asynccnt 0" ::: "memory");
#endif
    }
#else
    for (int idx = tid; idx < M_DIM * N_DIM; idx += 256) {
        int r = idx >> 7, c = idx & 127;
        Ash[r * A_PITCH + c] = Ab[idx];
    }
#endif
    if (tid < N_DIM) { xr[tid] = xraw_g[(size_t)t * N_DIM + tid]; zx[tid] = 0.f; }
    if (tid < M_DIM) {
        float bb = b_g[(size_t)t * M_DIM + tid];
        bv[tid] = bb; zs[tid] = 0.f;
        lam1[tid] = 1.f; lam2[tid] = 1.f;
        s1[tid] = fmaxf(bb, 1.f);   // s0 = max(h,1), h=[b;0]
        s2[tid] = 1.f;
    }
    __syncthreads();

    for (int it = 0; it < NITER; ++it) {
        // ---- residuals (structure-exploited) ----
        if (tid < M_DIM) {
            float acc = 0.f;
            const float* Ar = &Ash[tid * A_PITCH];
            for (int k = 0; k < N_DIM; ++k) acc = fmaf(Ar[k], zx[k], acc);
            rp1[tid] = acc - zs[tid] + s1[tid] - bv[tid];   // A zx - zs + s1 - b
            rp2[tid] = s2[tid] - zs[tid];                   // -zs + s2
            rds[tid] = zs[tid] - lam1[tid] - lam2[tid];     // c*zs - (lam1+lam2), c=1
        }
        // ---- mu = mean(lam*s): wave shuffle + 8-way cross-wave combine ----
        float c0 = (tid < 64) ? lam1[tid] * s1[tid]
                 : (tid < 128) ? lam2[tid - 64] * s2[tid - 64] : 0.f;
        for (int off = 16; off > 0; off >>= 1) c0 += __shfl_down(c0, off);
        if (lane == 0) red[wave] = c0;
        __syncthreads();
        if (tid == 0) {
            float s = 0.f;
            for (int w = 0; w < 8; ++w) s += red[w];
            scal[0] = SIGMA_C * s * (1.f / 128.f);
        }
        __syncthreads();
        const float sigmu = scal[0];

        if (tid < M_DIM) {
            float l1 = lam1[tid], l2 = lam2[tid], ss1 = s1[tid], ss2 = s2[tid];
            float r1 = sigmu - l1 * ss1, r2 = sigmu - l2 * ss2;  // r_cent
            rc1[tid] = r1; rc2[tid] = r2;
            float ww1 = l1 / ss1, ww2 = l2 / ss2;
            w1v[tid] = ww1;
            float qq1 = (r1 + l1 * rp1[tid]) / ss1;
            float qq2 = (r2 + l2 * rp2[tid]) / ss2;
            float rs  = -(rds[tid] - qq1 - qq2);                 // rhs_s
            rhss[tid] = rs;
            float dd  = 1.f + ww1 + ww2;                         // D = c + w1 + w2
            Dv[tid]  = dd;
            whv[tid] = ww1 * (1.f + ww2) / dd;                   // W-hat
            g1[tid]  = l1 + qq1 - ww1 * rs / dd;
        }
        __syncthreads();

        // rhat = x_raw - zx - A^T g1  (folds r_dual_x, rhs_x and D-elimination)
        if (tid < N_DIM) {
            float acc = 0.f;
            for (int c = 0; c < M_DIM; ++c) acc = fmaf(Ash[c * A_PITCH + tid], g1[c], acc);
            rhat[tid] = xr[tid] - zx[tid] - acc;
        }
        __syncthreads();

        // ---- S = A A^T via V_WMMA_F32_16X16X4_F32 (all 8 waves, EXEC all-1s) ----
        {
            const int ml = lane & 15;     // row within 16-block
            const int kh = lane >> 4;     // selects K pair {0,1} vs {2,3}
            for (int tt = 0; tt < 2; ++tt) {
                int tile = wave * 2 + tt;        // 16 tiles of 16x16
                int ti = tile >> 2, tj = tile & 3;
                const float* Ai = &Ash[(ti * 16 + ml) * A_PITCH + kh * 2];
                const float* Aj = &Ash[(tj * 16 + ml) * A_PITCH + kh * 2];
                v8f acc = {};
                for (int kk = 0; kk < N_DIM; kk += 4) {
                    v2f a, b2;
                    a.x  = Ai[kk]; a.y  = Ai[kk + 1];
                    b2.x = Aj[kk]; b2.y = Aj[kk + 1];
                    acc = __builtin_amdgcn_wmma_f32_16x16x4_f32(
                        false, a, false, b2, (short)0, acc, false, false);
                }
#pragma unroll
                for (int r = 0; r < 8; ++r) {
                    int mrow = r + kh * 8;       // C layout: vgpr r, lanes>=16 -> M=r+8
                    Ssh[(ti * 16 + mrow) * S_PITCH + tj * 16 + ml] = acc[r];
                }
            }
        }
        __syncthreads();
        if (tid < M_DIM) {
            Ssh[tid * S_PITCH + tid] += 1.f / whv[tid];          // + diag(1/w-hat)
            float acc = 0.f;                                     // u = A rhat
            const float* Ar = &Ash[tid * A_PITCH];
            for (int k = 0; k < N_DIM; ++k) acc = fmaf(Ar[k], rhat[k], acc);
            yv[tid] = acc;
        }
        __syncthreads();

        // ---- LDL^T factorization: ONE barrier per column, no sqrt ----
        // Column k kept unscaled (= L[:,k]*d[k]); full symmetric trailing update.
        for (int k = 0; k < M_DIM - 1; ++k) {
            float rdk = 1.f / Ssh[k * S_PITCH + k];              // broadcast + rcp
            int   j   = tid & 63;
            float cjk = Ssh[j * S_PITCH + k];                    // column element
            for (int i = k + 1 + (tid >> 6); i < M_DIM; i += 4) {
                if (j > k)
                    Ssh[i * S_PITCH + j] -= Ssh[i * S_PITCH + k] * cjk * rdk;
            }
            __syncthreads();
        }

        // ---- solves L y = u, D z = y, L^T x = z : single wave, zero barriers,
        //      vector in registers (2 rows/lane), pivots via v_readlane ----
        if (wave == 0) {
            float ya  = yv[lane], yb = yv[lane + 32];
            float rda = 1.f / Ssh[lane * S_PITCH + lane];
            float rdb = 1.f / Ssh[(lane + 32) * S_PITCH + (lane + 32)];
            for (int k = 0; k < M_DIM; ++k) {                    // forward (unit L)
                float yk  = (k < 32) ? __shfl(ya, k)  : __shfl(yb, k - 32);
                float rdk = (k < 32) ? __shfl(rda, k) : __shfl(rdb, k - 32);
                float tt  = yk * rdk;                            // L[i][k] = S[i][k]/d[k]
                if (lane > k)      ya -= Ssh[lane * S_PITCH + k] * tt;
                if (lane + 32 > k) yb -= Ssh[(lane + 32) * S_PITCH + k] * tt;
            }
            ya *= rda; yb *= rdb;                                // diagonal solve
            for (int k = M_DIM - 1; k >= 0; --k) {               // backward (unit L^T)
                float xk = (k < 32) ? __shfl(ya, k) : __shfl(yb, k - 32);
                if (lane < k)      ya -= Ssh[k * S_PITCH + lane] * rda * xk;
                if (lane + 32 < k) yb -= Ssh[k * S_PITCH + lane + 32] * rdb * xk;
            }
            yv[lane] = ya; yv[lane + 32] = yb;
        }
        __syncthreads();

        // dzx = rhat - A^T y   (Woodbury)
        if (tid < N_DIM) {
            float acc = 0.f;
            for (int c = 0; c < M_DIM; ++c) acc = fmaf(Ash[c * A_PITCH + tid], yv[c], acc);
            dzx[tid] = rhat[tid] - acc;
        }
        __syncthreads();
        if (tid < M_DIM) {
            float acc = 0.f;
            const float* Ar = &Ash[tid * A_PITCH];
            for (int k = 0; k < N_DIM; ++k) acc = fmaf(Ar[k], dzx[k], acc);  // A dzx
            float dzsv = (rhss[tid] + w1v[tid] * acc) / Dv[tid];
            dzs[tid] = dzsv;
            float dds1 = -rp1[tid] - (acc - dzsv);
            float dds2 = -rp2[tid] + dzsv;
            ds1v[tid] = dds1; ds2v[tid] = dds2;
            dl1[tid] = (rc1[tid] - lam1[tid] * dds1) / s1[tid];
            dl2[tid] = (rc2[tid] - lam2[tid] * dds2) / s2[tid];
        }
        __syncthreads();

        // ---- fraction-to-boundary alpha (shuffle min-reduce) ----
        float v, dv;
        if (tid < 64)       { v = lam1[tid];       dv = dl1[tid]; }
        else if (tid < 128) { v = lam2[tid - 64];  dv = dl2[tid - 64]; }
        else if (tid < 192) { v = s1[tid - 128];   dv = ds1v[tid - 128]; }
        else                { v = s2[tid - 192];   dv = ds2v[tid - 192]; }
        float cand = (dv < 0.f) ? (-v / dv) : 1e9f;
        for (int off = 16; off > 0; off >>= 1)
            cand = fminf(cand, __shfl_down(cand, off));
        if (lane == 0) red[wave] = cand;
        __syncthreads();
        if (tid == 0) {
            float mn = 1e9f;
            for (int w = 0; w < 8; ++w) mn = fminf(mn, red[w]);
            scal[1] = STEPF * fminf(1.f, mn);
        }
        __syncthreads();
        const float alpha = scal[1];

        if (tid < N_DIM) zx[tid] += alpha * dzx[tid];
        if (tid < M_DIM) {
            zs[tid]   += alpha * dzs[tid];
            lam1[tid] += alpha * dl1[tid];
            lam2[tid] += alpha * dl2[tid];
            s1[tid]   += alpha * ds1v[tid];
            s2[tid]   += alpha * ds2v[tid];
        }
        __syncthreads();
    }

    if (tid < N_DIM) out_g[(size_t)t * N_DIM + tid] = zx[tid];
}

extern "C" void kernel_launch(void* const* d_in, const int* in_sizes, int n_in,
                              void* d_out, int out_size, void* d_ws, size_t ws_size,
                              hipStream_t stream) {
    const float* xraw = (const float*)d_in[0];
    const float* A    = (const float*)d_in[1];
    const float* b    = (const float*)d_in[2];
    float* out        = (float*)d_out;
    const int T = in_sizes[0] / N_DIM;   // B*S = 2048
    qp_ipm_gfx1250_kernel<<<dim3(T), dim3(256), 0, stream>>>(xraw, A, b, out);
}